// GCNN_86277303042220
// MI455X (gfx1250) — compile-verified
//
#include <hip/hip_runtime.h>

#define NN 100000
#define FD 64
#define NC 2
#define NEG_SLOPE 0.2f

typedef __attribute__((ext_vector_type(2))) float v2f;
typedef __attribute__((ext_vector_type(8))) float v8f;

// Order-preserving float->uint encoding for atomic max (uint 0 == -NaN, safe
// identity since every node has a self-loop edge overwriting it).
__device__ __forceinline__ unsigned enc_f32(float f) {
    unsigned u = __float_as_uint(f);
    return (u & 0x80000000u) ? ~u : (u | 0x80000000u);
}
__device__ __forceinline__ float dec_f32(unsigned u) {
    return __uint_as_float((u & 0x80000000u) ? (u & 0x7FFFFFFFu) : ~u);
}

// Zero the aggregation accumulator, segment-max and segment-sum buffers.
__global__ void __launch_bounds__(256)
gat_init(float* __restrict__ acc, unsigned* __restrict__ mx,
         float* __restrict__ den, int n, int nf) {
    int i = blockIdx.x * 256 + threadIdx.x;
    if (i < nf) acc[i] = 0.0f;
    if (i < n) { mx[i] = 0u; den[i] = 0.0f; }
}

// H = X @ W  (fp32 WMMA 16x16x4), fused per-node attention logits
// asrc[i] = H[i,:] . a_s   adst[i] = H[i,:] . a_d
// One wave per 16-row tile. W is staged in LDS interleaved by K-pairs so each
// lane's B fragment {W[k][col], W[k+1][col]} is one contiguous ds_load_b64
// landing directly in an aligned VGPR pair (no 2addr loads / mov repacking).
__global__ void __launch_bounds__(256)
gat_gemm_attn(const float* __restrict__ X, const float* __restrict__ W,
              const float* __restrict__ a_s, const float* __restrict__ a_d,
              float* __restrict__ H, float* __restrict__ asrc,
              float* __restrict__ adst, int nTiles) {
    __shared__ float Wl[FD * FD];   // [pair p][col j][k&1] = W[2p + (k&1)][j]
    __shared__ float asl[FD];
    __shared__ float adl[FD];
    for (int i = threadIdx.x; i < FD * FD; i += 256) {
        int k = i >> 6, j = i & 63;
        Wl[(k >> 1) * (2 * FD) + j * 2 + (k & 1)] = W[i];
    }
    if (threadIdx.x < FD) asl[threadIdx.x] = a_s[threadIdx.x];
    else if (threadIdx.x < 2 * FD) adl[threadIdx.x - FD] = a_d[threadIdx.x - FD];
    __syncthreads();

    int wave = threadIdx.x >> 5;
    int lane = threadIdx.x & 31;
    int tile = blockIdx.x * 8 + wave;
    if (tile >= nTiles) return;

    int row0 = tile * 16;
    int col = lane & 15;        // N position (B/C/D) and M row (A)
    int hi  = lane >> 4;        // lane half selects K pair / M+8
    int arow = row0 + col;

    v8f c0 = {}, c1 = {}, c2 = {}, c3 = {};
    #pragma unroll
    for (int k0 = 0; k0 < FD; k0 += 4) {
        // A 16x4: lane holds A[row][k0+2*hi], A[row][k0+2*hi+1] (one b64 load)
        v2f a = *(const v2f*)(X + (size_t)arow * FD + k0 + 2 * hi);
        // B 4x16: pair index p = (k0 + 2*hi)/2; contiguous float2 per lane
        const v2f* bp = (const v2f*)(Wl + (k0 / 2 + hi) * (2 * FD));
        v2f b0 = bp[col];
        v2f b1 = bp[col + 16];
        v2f b2 = bp[col + 32];
        v2f b3 = bp[col + 48];
        c0 = __builtin_amdgcn_wmma_f32_16x16x4_f32(false, a, false, b0, (short)0, c0, false, false);
        c1 = __builtin_amdgcn_wmma_f32_16x16x4_f32(false, a, false, b1, (short)0, c1, false, false);
        c2 = __builtin_amdgcn_wmma_f32_16x16x4_f32(false, a, false, b2, (short)0, c2, false, false);
        c3 = __builtin_amdgcn_wmma_f32_16x16x4_f32(false, a, false, b3, (short)0, c3, false, false);
    }

    // Write H tile + accumulate attention-logit partials.
    float pa[8], pd[8];
    #pragma unroll
    for (int r = 0; r < 8; ++r) {
        int m = row0 + r + 8 * hi;
        float h0 = c0[r], h1 = c1[r], h2 = c2[r], h3 = c3[r];
        float* hp = H + (size_t)m * FD;
        hp[col] = h0; hp[col + 16] = h1; hp[col + 32] = h2; hp[col + 48] = h3;
        pa[r] = h0 * asl[col] + h1 * asl[col + 16] + h2 * asl[col + 32] + h3 * asl[col + 48];
        pd[r] = h0 * adl[col] + h1 * adl[col + 16] + h2 * adl[col + 32] + h3 * adl[col + 48];
    }
    // Reduce across the 16 lanes holding the same row (masks < 16 stay in-half).
    #pragma unroll
    for (int msk = 1; msk < 16; msk <<= 1) {
        #pragma unroll
        for (int r = 0; r < 8; ++r) {
            pa[r] += __shfl_xor(pa[r], msk, 32);
            pd[r] += __shfl_xor(pd[r], msk, 32);
        }
    }
    if (col == 0) {
        #pragma unroll
        for (int r = 0; r < 8; ++r) {
            asrc[row0 + r + 8 * hi] = pa[r];
            adst[row0 + r + 8 * hi] = pd[r];
        }
    }
}

// Per edge (incl. implicit self-loops at idx >= E): leaky-relu logit + seg-max.
__global__ void __launch_bounds__(256)
gat_edge_logits(const int* __restrict__ ei, const float* __restrict__ asrc,
                const float* __restrict__ adst, float* __restrict__ lg,
                unsigned* __restrict__ mx, int E, int n) {
    int i = blockIdx.x * 256 + threadIdx.x;
    if (i >= E + n) return;
    int s, d;
    if (i < E) { s = ei[i]; d = ei[E + i]; } else { s = d = i - E; }
    float l = asrc[s] + adst[d];
    l = l > 0.0f ? l : NEG_SLOPE * l;
    lg[i] = l;
    atomicMax(mx + d, enc_f32(l));
}

// Per edge: ex = exp(l - max[dst]); seg-sum denominator. lg buffer becomes ex.
__global__ void __launch_bounds__(256)
gat_edge_exp(const int* __restrict__ ei, const unsigned* __restrict__ mx,
             float* __restrict__ lg, float* __restrict__ den, int E, int n) {
    int i = blockIdx.x * 256 + threadIdx.x;
    if (i >= E + n) return;
    int d = (i < E) ? ei[E + i] : (i - E);
    float e = __expf(lg[i] - dec_f32(mx[d]));
    lg[i] = e;
    atomicAdd(den + d, e);
}

// 64 threads per edge: coalesced gather of h[src], fp32 atomic scatter to acc[dst].
__global__ void __launch_bounds__(256)
gat_scatter(const int* __restrict__ ei, const float* __restrict__ ex,
            const float* __restrict__ den, const float* __restrict__ H,
            float* __restrict__ acc, int E, int n) {
    int e = blockIdx.x * 4 + (threadIdx.x >> 6);
    int f = threadIdx.x & 63;
    if (e >= E + n) return;
    int s, d;
    if (e < E) { s = ei[e]; d = ei[E + e]; } else { s = d = e - E; }
    float alpha = ex[e] / (den[d] + 1e-16f);
    atomicAdd(acc + (size_t)d * FD + f, alpha * H[(size_t)s * FD + f]);
}

// In-place out = relu(acc + b)
__global__ void __launch_bounds__(256)
gat_bias_relu(float* __restrict__ acc, const float* __restrict__ b, int nf) {
    int i = blockIdx.x * 256 + threadIdx.x;
    if (i >= nf) return;
    float v = acc[i] + b[i & (FD - 1)];
    acc[i] = v > 0.0f ? v : 0.0f;
}

// Wave per node: logits = h @ Wc + bc, then log_softmax over 2 classes.
__global__ void __launch_bounds__(256)
gat_classifier(const float* __restrict__ H, const float* __restrict__ Wc,
               const float* __restrict__ bc, float* __restrict__ out, int n) {
    int wave = threadIdx.x >> 5;
    int lane = threadIdx.x & 31;
    int node = blockIdx.x * 8 + wave;
    if (node >= n) return;
    v2f h = *(const v2f*)(H + (size_t)node * FD + 2 * lane);
    float z0 = h.x * Wc[(2 * lane) * NC + 0] + h.y * Wc[(2 * lane + 1) * NC + 0];
    float z1 = h.x * Wc[(2 * lane) * NC + 1] + h.y * Wc[(2 * lane + 1) * NC + 1];
    #pragma unroll
    for (int msk = 1; msk < 32; msk <<= 1) {
        z0 += __shfl_xor(z0, msk, 32);
        z1 += __shfl_xor(z1, msk, 32);
    }
    if (lane == 0) {
        z0 += bc[0]; z1 += bc[1];
        float m = fmaxf(z0, z1);
        float lse = m + __logf(__expf(z0 - m) + __expf(z1 - m));
        out[node * 2 + 0] = z0 - lse;
        out[node * 2 + 1] = z1 - lse;
    }
}

extern "C" void kernel_launch(void* const* d_in, const int* in_sizes, int n_in,
                              void* d_out, int out_size, void* d_ws, size_t ws_size,
                              hipStream_t stream) {
    const float* x   = (const float*)d_in[0];
    const int*   ei  = (const int*)d_in[1];
    const float* W1  = (const float*)d_in[2];
    const float* as1 = (const float*)d_in[3];
    const float* ad1 = (const float*)d_in[4];
    const float* b1  = (const float*)d_in[5];
    const float* W2  = (const float*)d_in[6];
    const float* as2 = (const float*)d_in[7];
    const float* ad2 = (const float*)d_in[8];
    const float* b2  = (const float*)d_in[9];
    const float* Wc  = (const float*)d_in[10];
    const float* bc  = (const float*)d_in[11];

    const int n = in_sizes[0] / FD;      // 100000
    const int E = in_sizes[1] / 2;       // 1200000

    // Workspace carve-up (floats): H | acc1 | acc2 | asrc | adst | mx | den | ex
    float* ws   = (float*)d_ws;
    float* H    = ws;
    float* acc1 = H    + (size_t)n * FD;
    float* acc2 = acc1 + (size_t)n * FD;
    float* asrc = acc2 + (size_t)n * FD;
    float* adst = asrc + n;
    unsigned* mx = (unsigned*)(adst + n);
    float* den  = (float*)(mx + n);
    float* ex   = den + n;               // E + n entries

    const int nTiles = (n + 15) / 16;
    dim3 blk(256);
    const int gInit = (n * FD + 255) / 256;
    const int gGemm = (nTiles + 7) / 8;
    const int gEdge = (E + n + 255) / 256;
    const int gScat = (E + n + 3) / 4;
    const int gCls  = (n + 7) / 8;

    // ---- layer 1 ----
    gat_init       <<<gInit, blk, 0, stream>>>(acc1, mx, den, n, n * FD);
    gat_gemm_attn  <<<gGemm, blk, 0, stream>>>(x, W1, as1, ad1, H, asrc, adst, nTiles);
    gat_edge_logits<<<gEdge, blk, 0, stream>>>(ei, asrc, adst, ex, mx, E, n);
    gat_edge_exp   <<<gEdge, blk, 0, stream>>>(ei, mx, ex, den, E, n);
    gat_scatter    <<<gScat, blk, 0, stream>>>(ei, ex, den, H, acc1, E, n);
    gat_bias_relu  <<<gInit, blk, 0, stream>>>(acc1, b1, n * FD);

    // ---- layer 2 ----
    gat_init       <<<gInit, blk, 0, stream>>>(acc2, mx, den, n, n * FD);
    gat_gemm_attn  <<<gGemm, blk, 0, stream>>>(acc1, W2, as2, ad2, H, asrc, adst, nTiles);
    gat_edge_logits<<<gEdge, blk, 0, stream>>>(ei, asrc, adst, ex, mx, E, n);
    gat_edge_exp   <<<gEdge, blk, 0, stream>>>(ei, mx, ex, den, E, n);
    gat_scatter    <<<gScat, blk, 0, stream>>>(ei, ex, den, H, acc2, E, n);
    gat_bias_relu  <<<gInit, blk, 0, stream>>>(acc2, b2, n * FD);

    // ---- classifier + log_softmax ----
    gat_classifier <<<gCls, blk, 0, stream>>>(acc2, Wc, bc, (float*)d_out, n);
}